// TensorProductTorch_12335146074312
// MI455X (gfx1250) — compile-verified
//
#include <hip/hip_runtime.h>

typedef __attribute__((ext_vector_type(2))) float v2f;
typedef __attribute__((ext_vector_type(8))) float v8f;

// ---- Clebsch-Gordan derived constants (component normalization) ----
#define IS2 0.70710678118654752f  // 1/sqrt(2)
#define IS3 0.57735026918962576f  // 1/sqrt(3)
#define IS5 0.44721359549995794f  // 1/sqrt(5)
#define IS6 0.40824829046386302f  // 1/sqrt(6)
#define HLF 0.86602540378443865f  // sqrt(3)/2
#define G25 0.63245553203367587f  // sqrt(2/5)
#define G1C 0.53452248382484879f  // sqrt(2/7)
#define G2C 0.26726124191242439f  // sqrt(1/14)
#define G3C 0.46291004988627573f  // sqrt(3/14)

// x1 per-u feature f (0..8): [A, Bv0..2, Cq0..4]; f>=9 is K-padding (zero)
__device__ __forceinline__ float x1_feat(const float* __restrict__ x1r, int u, int f) {
  if (f == 0) return x1r[u];                       // l=0 block
  if (f < 4)  return x1r[128 + 3 * u + (f - 1)];   // l=1 block
  if (f < 9)  return x1r[512 + 5 * u + (f - 4)];   // l=2 block
  return 0.0f;
}

__global__ __launch_bounds__(256) void e3nn_tp_wmma(
    const float* __restrict__ x1, const float* __restrict__ x2,
    float* __restrict__ out, int B) {
  const int b = blockIdx.x;
  if (b >= B) return;
  const int tid = threadIdx.x;

  // Per-row mixing matrix M[12][48] (K padded 9->12, N padded 35->48)
  __shared__ float Ml[12 * 48];

  if (tid < 12) {
    const float* x2r = x2 + (size_t)b * 9;
    const float S0 = x2r[0];
    const float v0 = x2r[1], v1 = x2r[2], v2 = x2r[3];            // l=1: (y,z,x)
    const float q0 = x2r[4], q1 = x2r[5], q2 = x2r[6], q3 = x2r[7], q4 = x2r[8];
    float* R = &Ml[tid * 48];
#pragma unroll
    for (int c = 0; c < 48; ++c) R[c] = 0.0f;
    switch (tid) {
      case 0:  // feature A (l=0 of x1)
        R[0] = S0;                                  // (0,0->0)
        R[3] = v0; R[4] = v1; R[5] = v2;            // (0,1->1)
        R[15] = q0; R[16] = q1; R[17] = q2; R[18] = q3; R[19] = q4;  // (0,2->2)
        break;
      case 1:  // Bv[0] (y)
        R[1] = v0 * IS3;                            // (1,1->0)
        R[6] = S0;                                  // (1,0->1)
        R[9]  = G25 * (-0.5f * q2 - HLF * q4);      // (1,2->1)
        R[10] = G25 * HLF * q1;
        R[11] = G25 * HLF * q0;
        R[20] = v2 * IS2; R[21] = v1 * IS2;         // (1,1->2)
        R[22] = -v0 * IS6; R[24] = -v0 * IS2;
        break;
      case 2:  // Bv[1] (z)
        R[1] = v1 * IS3;
        R[7] = S0;
        R[9]  = G25 * HLF * q1;
        R[10] = G25 * q2;
        R[11] = G25 * HLF * q3;
        R[21] = v0 * IS2; R[22] = 2.0f * v1 * IS6; R[23] = v2 * IS2;
        break;
      case 3:  // Bv[2] (x)
        R[1] = v2 * IS3;
        R[8] = S0;
        R[9]  = G25 * HLF * q0;
        R[10] = G25 * HLF * q3;
        R[11] = G25 * (-0.5f * q2 + HLF * q4);
        R[20] = v0 * IS2; R[22] = -v2 * IS6; R[23] = v1 * IS2; R[24] = v2 * IS2;
        break;
      case 4:  // Cq[0] (xy)
        R[2] = q0 * IS5;                            // (2,2->0)
        R[12] = G25 * HLF * v2; R[14] = G25 * HLF * v0;  // (2,1->1)
        R[25] = S0;                                 // (2,0->2)
        R[30] = -G1C * q2; R[31] = G3C * q3;        // (2,2->2)
        R[32] = -G1C * q0; R[33] = G3C * q1;
        break;
      case 5:  // Cq[1] (yz)
        R[2] = q1 * IS5;
        R[12] = G25 * HLF * v1; R[13] = G25 * HLF * v0;
        R[26] = S0;
        R[30] = G3C * q3; R[31] = G2C * q2 - G3C * q4;
        R[32] = G2C * q1; R[33] = G3C * q0; R[34] = -G3C * q1;
        break;
      case 6:  // Cq[2] (z^2)
        R[2] = q2 * IS5;
        R[12] = -0.5f * G25 * v0; R[13] = G25 * v1; R[14] = -0.5f * G25 * v2;
        R[27] = S0;
        R[30] = -G1C * q0; R[31] = G2C * q1; R[32] = G1C * q2;
        R[33] = G2C * q3; R[34] = -G1C * q4;
        break;
      case 7:  // Cq[3] (zx)
        R[2] = q3 * IS5;
        R[13] = G25 * HLF * v2; R[14] = G25 * HLF * v1;
        R[28] = S0;
        R[30] = G3C * q1; R[31] = G3C * q0; R[32] = G2C * q3;
        R[33] = G2C * q2 + G3C * q4; R[34] = G3C * q3;
        break;
      case 8:  // Cq[4] (x^2-y^2)
        R[2] = q4 * IS5;
        R[12] = -G25 * HLF * v0; R[14] = G25 * HLF * v2;
        R[29] = S0;
        R[31] = -G3C * q1; R[32] = -G1C * q4; R[33] = G3C * q3; R[34] = -G1C * q2;
        break;
      default: break;  // rows 9..11: K padding, stay zero
    }
  }
  __syncthreads();

  const int lane = tid & 31;
  const int wave = tid >> 5;   // 8 waves, each owns 16 u-rows
  const int hi   = lane >> 4;  // lane half selects K pair {2hi, 2hi+1} (+4t)
  const int lo   = lane & 15;
  const int u0   = wave << 4;
  const int u    = u0 + lo;
  const float* x1r = x1 + (size_t)b * 1152;

  // A fragments: lane holds A[M=lo][K=4t+2hi+r] for t=0..2, r=0..1
  float fa[6];
#pragma unroll
  for (int t = 0; t < 3; ++t)
#pragma unroll
    for (int r = 0; r < 2; ++r)
      fa[2 * t + r] = x1_feat(x1r, u, 4 * t + 2 * hi + r);

  v8f acc[3] = {};

#if __has_builtin(__builtin_amdgcn_wmma_f32_16x16x4_f32)
#pragma unroll
  for (int t = 0; t < 3; ++t) {
    v2f a;
    a.x = fa[2 * t];
    a.y = fa[2 * t + 1];
    const int kk = 4 * t + 2 * hi;
#pragma unroll
    for (int n = 0; n < 3; ++n) {
      v2f bm;
      bm.x = Ml[kk * 48 + n * 16 + lo];
      bm.y = Ml[(kk + 1) * 48 + n * 16 + lo];
      acc[n] = __builtin_amdgcn_wmma_f32_16x16x4_f32(
          false, a, false, bm, (short)0, acc[n], false, false);
    }
  }
#else
  // VALU fallback with identical C-fragment layout
#pragma unroll
  for (int n = 0; n < 3; ++n)
#pragma unroll
    for (int j = 0; j < 8; ++j) {
      const int uu = u0 + 8 * hi + j;
      const int c  = n * 16 + lo;
      float s = 0.0f;
#pragma unroll
      for (int f = 0; f < 9; ++f) s += x1_feat(x1r, uu, f) * Ml[f * 48 + c];
      acc[n][j] = s;
    }
#endif

  // Scatter C fragments to the sorted e3nn output layout.
  // Columns: 0..2 -> l=0 chunks (width 1), 3..14 -> l=1 chunks (width 3),
  // 15..34 -> l=2 chunks (width 5); 35..47 are N padding (dropped).
  float* outr = out + (size_t)b * 4480;
  const int ustart = u0 + 8 * hi;  // C layout: lanes>=16 hold rows M=8..15
#pragma unroll
  for (int n = 0; n < 3; ++n) {
    const int c = n * 16 + lo;
    if (c < 35) {
      int base, width, kk;
      if (c < 3)       { base = c * 128;                     width = 1; kk = 0; }
      else if (c < 15) { int t2 = c - 3;  base = 384  + (t2 / 3) * 384; width = 3; kk = t2 % 3; }
      else             { int t2 = c - 15; base = 1920 + (t2 / 5) * 640; width = 5; kk = t2 % 5; }
      float* p = outr + base + ustart * width + kk;
#pragma unroll
      for (int j = 0; j < 8; ++j) p[j * width] = acc[n][j];
    }
  }
}

extern "C" void kernel_launch(void* const* d_in, const int* in_sizes, int n_in,
                              void* d_out, int out_size, void* d_ws, size_t ws_size,
                              hipStream_t stream) {
  const float* x1 = (const float*)d_in[0];
  const float* x2 = (const float*)d_in[1];
  float* out = (float*)d_out;
  const int B = in_sizes[1] / 9;  // input2 is (B, 9)
  e3nn_tp_wmma<<<dim3((unsigned)B), dim3(256), 0, stream>>>(x1, x2, out, B);
  (void)n_in; (void)out_size; (void)d_ws; (void)ws_size;
}